// RandomPooling_9242769621166
// MI455X (gfx1250) — compile-verified
//
#include <hip/hip_runtime.h>

// RandomPooling: y[b,c,h,w] = x[b, c, reflect(h+rh-1), reflect(w+rw-1)]
// B=32, C=3, H=W=512, GAP=1. Pure gather: 0 FLOPs, ~268 MB of traffic
// -> HBM-bandwidth bound (~11.5 us at 23.3 TB/s). Optimization targets:
//   * NT hints on streaming output/indices so x (100 MB) stays in 192 MB L2
//   * gfx1250 global_prefetch_b8 of the 3x3-row source window to overlap
//     the index-load -> gather dependency chain
//   * SGPR-uniform row bases (block == one (b,h) row), 128-bit vector ops

#define NB 32
#define NC 3
#define NH 512
#define NW 512

// Native clang vector types (required by __builtin_nontemporal_*; HIP's
// int4/float4 are class wrappers and are rejected by the builtin).
typedef int   v4i __attribute__((ext_vector_type(4)));
typedef float v4f __attribute__((ext_vector_type(4)));

__global__ __launch_bounds__(128) void random_pool_kernel(
    const float* __restrict__ x,
    const int*   __restrict__ rh,
    const int*   __restrict__ rw,
    float*       __restrict__ out)
{
    // One block = one (b, h) row; 128 threads x 4 pixels = 512 = W.
    const int b  = blockIdx.x >> 9;        // blockIdx.x / NH
    const int h  = blockIdx.x & (NH - 1);  // blockIdx.x % NH
    const int w4 = threadIdx.x << 2;       // first of 4 pixels

    // Candidate source rows (reflect at borders): h-1, h, h+1
    const int hm = (h - 1 < 0)   ? 1      : h - 1;
    const int hp = (h + 1 >= NH) ? NH - 2 : h + 1;

    const float* __restrict__ xb = x + (size_t)b * (NC * NH * NW); // image base (SGPR)

    // Prefetch the 3-row x 3-channel source window (independent of the index
    // loads below) -> gfx1250 global_prefetch_b8; no LOADcnt cost.
#pragma unroll
    for (int c = 0; c < NC; ++c) {
        const int plane = c * NH * NW;
        __builtin_prefetch(xb + plane + hm * NW + w4, 0, 3);
        __builtin_prefetch(xb + plane + h  * NW + w4, 0, 3);
        __builtin_prefetch(xb + plane + hp * NW + w4, 0, 3);
    }

    // Indices: read-once -> non-temporal 128-bit loads.
    const int idxBase = (b * NH + h) * NW + w4;
    const v4i rh4 = __builtin_nontemporal_load((const v4i*)(rh + idxBase));
    const v4i rw4 = __builtin_nontemporal_load((const v4i*)(rw + idxBase));

    const int rhv[4] = {rh4.x, rh4.y, rh4.z, rh4.w};
    const int rwv[4] = {rw4.x, rw4.y, rw4.z, rw4.w};

    int soff[4]; // per-pixel source offset within a channel plane
#pragma unroll
    for (int j = 0; j < 4; ++j) {
        int sh = h + rhv[j] - 1;                  // in [-1, H]
        sh = (sh < 0) ? -sh : sh;                 // -1 -> 1
        sh = (sh >= NH) ? (2 * NH - 2 - sh) : sh; // H -> H-2
        int sw = w4 + j + rwv[j] - 1;             // in [-1, W]
        sw = (sw < 0) ? -sw : sw;
        sw = (sw >= NW) ? (2 * NW - 2 - sw) : sw;
        soff[j] = sh * NW + sw;
    }

    float* __restrict__ ob = out + (size_t)b * (NC * NH * NW) + (size_t)h * NW + w4;

#pragma unroll
    for (int c = 0; c < NC; ++c) {
        const int plane = c * NH * NW;
        v4f v;
        v.x = xb[plane + soff[0]];
        v.y = xb[plane + soff[1]];
        v.z = xb[plane + soff[2]];
        v.w = xb[plane + soff[3]];
        // Output is write-once: non-temporal so it doesn't evict x from L2.
        __builtin_nontemporal_store(v, (v4f*)(ob + (size_t)plane));
    }
}

extern "C" void kernel_launch(void* const* d_in, const int* in_sizes, int n_in,
                              void* d_out, int out_size, void* d_ws, size_t ws_size,
                              hipStream_t stream)
{
    (void)in_sizes; (void)n_in; (void)out_size; (void)d_ws; (void)ws_size;
    const float* x  = (const float*)d_in[0];
    const int*   rh = (const int*)d_in[1];
    const int*   rw = (const int*)d_in[2];
    float*       y  = (float*)d_out;

    dim3 grid(NB * NH);  // 16384 blocks, each one (b,h) row
    dim3 block(128);     // 4 wave32 per block
    random_pool_kernel<<<grid, block, 0, stream>>>(x, rh, rw, y);
}